// TaskLoss_64269890618031
// MI455X (gfx1250) — compile-verified
//
#include <hip/hip_runtime.h>
#include <math.h>

// ---------------------------------------------------------------------------
// PPO policy loss on MI455X (gfx1250).
// Memory-bound: 256 MiB streamed once -> ~11 us floor @ 23.3 TB/s.
// Single-pass linear-recurrence scan: per-thread chunk is F(c) = B + P*c,
// moments accumulated with zero carry and corrected after a suffix-composition
// scan across the block. Rewards staged to LDS with CDNA5 async copies;
// final partial reduction uses V_WMMA_F32_16X16X4_F32 with an all-ones B.
// ---------------------------------------------------------------------------

#define TPB     256
#define S_LEN   8192
#define B_ROWS  2048
#define CHUNK   32          // S_LEN / TPB elements per thread
#define GAMMA_F 0.99f

typedef __attribute__((ext_vector_type(4))) float f4;
typedef __attribute__((ext_vector_type(4))) int   i4;
typedef __attribute__((ext_vector_type(2))) float v2f;
typedef __attribute__((ext_vector_type(8))) float v8f;

#if defined(__AMDGCN__) && __has_builtin(__builtin_amdgcn_global_load_async_to_lds_b64)
#define USE_ASYNC_LDS 1
// The builtin's parameters are v2i32 pointers in AS1 (global) / AS3 (LDS).
typedef int v2i_t __attribute__((vector_size(8)));
typedef __attribute__((address_space(1))) v2i_t* as1_v2i;
typedef __attribute__((address_space(3))) v2i_t* as3_v2i;
typedef __attribute__((address_space(1))) void* as1_void;
typedef __attribute__((address_space(3))) void* as3_void;
#else
#define USE_ASYNC_LDS 0
#endif

#if defined(__AMDGCN__) && __has_builtin(__builtin_amdgcn_wmma_f32_16x16x4_f32)
#define USE_WMMA_RED 1
#else
#define USE_WMMA_RED 0
#endif

// Padded LDS index: +2 floats per 32 elements. Per-wave read stride becomes
// 34 words -> 32 distinct banks (gcd(34,64)=2, injective over 32 lanes),
// while 8-byte (b64) granule alignment is preserved for the async copies.
__device__ __forceinline__ int pidx(int t) { return t + ((t >> 5) << 1); }

__global__ __launch_bounds__(TPB)
void ppo_row_kernel(const float* __restrict__ lp, const float* __restrict__ rw,
                    const float* __restrict__ val, const int* __restrict__ dn,
                    float* __restrict__ partials)
{
  __shared__ float  s_r[S_LEN + ((S_LEN >> 5) << 1)];  // padded rewards row
  __shared__ float  scA[TPB];                           // scan: P (slope)
  __shared__ float  scB[TPB];                           // scan: B (offset)
  __shared__ double sAcc[4][TPB / 32];

  const int tid = threadIdx.x;
  const long long base = (long long)blockIdx.x * S_LEN;
  const float* rwRow = rw + base;

  // ---- stage rewards row into LDS (coalesced 8B granules) ----
#if USE_ASYNC_LDS
  #pragma unroll
  for (int i = 0; i < 16; ++i) {
    const int g = tid + i * TPB;               // 8-byte granule index
    const int a = (g << 1) + ((g >> 4) << 1);  // padded float index (8B aligned)
    __builtin_amdgcn_global_load_async_to_lds_b64(
        (as1_v2i)(as1_void)(void*)(rwRow + (g << 1)),
        (as3_v2i)(as3_void)(void*)(s_r + a),
        0, 0);
  }
#if __has_builtin(__builtin_amdgcn_s_wait_asynccnt)
  __builtin_amdgcn_s_wait_asynccnt(0);
#else
  asm volatile("s_wait_asynccnt 0" ::: "memory");
#endif
#else
  for (int i = 0; i < 16; ++i) {
    const int g = tid + i * TPB;
    const int a = (g << 1) + ((g >> 4) << 1);
    const float* p = rwRow + (g << 1);
    s_r[a] = p[0]; s_r[a + 1] = p[1];
  }
#endif
  __syncthreads();

  // ---- per-thread backward chunk scan, single pass over all inputs ----
  // ret_t = B_t + P_t * c  where c = return entering from the right.
  const int t0 = tid * CHUNK;
  float Bv = 0.f, P = 1.f;
  float aB = 0.f, aBB = 0.f, aP = 0.f, aPP = 0.f;
  float aPB = 0.f, aLB = 0.f, aLP = 0.f, aL = 0.f;
  #pragma unroll
  for (int g = (CHUNK >> 2) - 1; g >= 0; --g) {
    const int tg = t0 + (g << 2);
    const f4 v4 = __builtin_nontemporal_load((const f4*)(val + base + tg));
    const f4 l4 = __builtin_nontemporal_load((const f4*)(lp  + base + tg));
    const i4 d4 = __builtin_nontemporal_load((const i4*)(dn  + base + tg));
    #pragma unroll
    for (int e = 3; e >= 0; --e) {
      const float r    = s_r[pidx(tg + e)];
      const float coef = (d4[e] != 0) ? 0.0f : GAMMA_F;   // gamma*(1-done)
      Bv = fmaf(coef, Bv, r);       // zero-carry return at t
      P  = coef * P;                // product of coefficients over suffix
      const float bb = Bv - v4[e];  // adv_t = bb + P*c
      const float l  = l4[e];
      aB  += bb;              aBB = fmaf(bb, bb, aBB);
      aP  += P;               aPP = fmaf(P, P, aPP);
      aPB = fmaf(P, bb, aPB); aLB = fmaf(l, bb, aLB);
      aLP = fmaf(l, P, aLP);  aL  += l;
    }
  }

  // ---- suffix composition scan over 256 chunk functions (Hillis-Steele) ----
  scA[tid] = P; scB[tid] = Bv;
  __syncthreads();
  #pragma unroll
  for (int d = 1; d < TPB; d <<= 1) {
    const bool ok = (tid + d) < TPB;
    const float a1 = scA[tid], b1 = scB[tid];
    float a2 = 0.f, b2 = 0.f;
    if (ok) { a2 = scA[tid + d]; b2 = scB[tid + d]; }
    __syncthreads();
    if (ok) { scA[tid] = a1 * a2; scB[tid] = fmaf(a1, b2, b1); }
    __syncthreads();
  }
  const float c = (tid + 1 < TPB) ? scB[tid + 1] : 0.0f;  // carry into chunk

  // ---- correct moments with true carry ----
  const float q0 = aB  + c * aP;                           // sum(adv)
  const float q1 = aBB + 2.0f * c * aPB + c * c * aPP;     // sum(adv^2)
  const float q2 = aLB + c * aLP;                          // sum(lp*adv)
  const float q3 = aL;                                     // sum(lp)

  // ---- deterministic block reduction (wave32 shuffles + LDS) ----
  double r0 = q0, r1 = q1, r2 = q2, r3 = q3;
  #pragma unroll
  for (int off = 16; off > 0; off >>= 1) {
    r0 += __shfl_down(r0, off);
    r1 += __shfl_down(r1, off);
    r2 += __shfl_down(r2, off);
    r3 += __shfl_down(r3, off);
  }
  const int lane = tid & 31, wv = tid >> 5;
  if (lane == 0) { sAcc[0][wv] = r0; sAcc[1][wv] = r1; sAcc[2][wv] = r2; sAcc[3][wv] = r3; }
  __syncthreads();
  if (tid < 4) {
    double s = 0.0;
    for (int w = 0; w < TPB / 32; ++w) s += sAcc[tid][w];
    partials[(size_t)blockIdx.x * 4 + tid] = (float)s;
  }
}

// ---------------------------------------------------------------------------
// Finalize: reduce 2048x4 partials and emit the scalar loss.
// WMMA path: D = A x ones + C with V_WMMA_F32_16X16X4_F32. Since B is all
// ones its lane layout is irrelevant; A/D layouts are the documented ones.
// Row m of A carries only quantity (m&3), so row sums never mix quantities.
// ---------------------------------------------------------------------------
__global__ __launch_bounds__(TPB)
void ppo_finalize_kernel(const float* __restrict__ partials, float* __restrict__ out)
{
  const int tid = threadIdx.x;
  __shared__ double wsum[4][TPB / 32];

#if USE_WMMA_RED
  const int lane = tid & 31;
  const int wv   = tid >> 5;
  const int lam  = lane & 15;   // A row index m
  const int hi   = lane >> 4;   // selects K pair {0,1} vs {2,3}
  const int q    = lam & 3;     // quantity carried by this row
  const int g    = lam >> 2;    // row group within quantity
  v8f acc = {};
  v2f ones; ones[0] = 1.0f; ones[1] = 1.0f;
  for (int k = 0; k < 16; ++k) {            // 8 waves x 16 = 128 tiles of 16 blocks
    const int s  = wv + (k << 3);
    const int i0 = (s << 4) + (g << 2) + (hi << 1);
    v2f a;
    a[0] = partials[(size_t)(i0 + 0) * 4 + q];
    a[1] = partials[(size_t)(i0 + 1) * 4 + q];
    // D[m][n] = sum_k A[m][k] + C[m][n]  (B == ones)
    acc = __builtin_amdgcn_wmma_f32_16x16x4_f32(false, a, false, ones,
                                                (short)0, acc, false, false);
  }
  // D layout: VGPR v, lanes 0-15 -> row v; lanes 16-31 -> row v+8 (replicated over n).
  #pragma unroll
  for (int qq = 0; qq < 4; ++qq) {
    const float u = acc[qq] + acc[qq + 4];          // rows qq,qq+4 (or +8,+12 in hi lanes)
    const float t = __shfl(u, 0) + __shfl(u, 16);   // add low-lane and high-lane rows
    if (lane == 0) wsum[qq][wv] = (double)t;
  }
  __syncthreads();
#else
  double a0 = 0, a1 = 0, a2 = 0, a3 = 0;
  for (int i = tid; i < B_ROWS; i += TPB) {
    const float* p = partials + (size_t)i * 4;
    a0 += p[0]; a1 += p[1]; a2 += p[2]; a3 += p[3];
  }
  #pragma unroll
  for (int off = 16; off > 0; off >>= 1) {
    a0 += __shfl_down(a0, off);
    a1 += __shfl_down(a1, off);
    a2 += __shfl_down(a2, off);
    a3 += __shfl_down(a3, off);
  }
  const int lane = tid & 31, wv = tid >> 5;
  if (lane == 0) { wsum[0][wv] = a0; wsum[1][wv] = a1; wsum[2][wv] = a2; wsum[3][wv] = a3; }
  __syncthreads();
#endif

  if (tid == 0) {
    double S1 = 0, S2 = 0, S3 = 0, S4 = 0;
    for (int w = 0; w < TPB / 32; ++w) {
      S1 += wsum[0][w]; S2 += wsum[1][w]; S3 += wsum[2][w]; S4 += wsum[3][w];
    }
    const double N    = (double)B_ROWS * (double)S_LEN;
    const double mean = S1 / N;
    double var = (S2 - S1 * S1 / N) / (N - 1.0);   // ddof=1
    if (var < 0.0) var = 0.0;
    const double sd   = sqrt(var);
    const double loss = -(S3 - mean * S4) / (N * (sd + 1e-8));
    out[0] = (float)loss;
  }
}

extern "C" void kernel_launch(void* const* d_in, const int* in_sizes, int n_in,
                              void* d_out, int out_size, void* d_ws, size_t ws_size,
                              hipStream_t stream) {
  const float* lp = (const float*)d_in[0];  // log_probs (B,S) f32
  const float* rw = (const float*)d_in[1];  // rewards   (B,S) f32
  const float* vl = (const float*)d_in[2];  // values    (B,S) f32
  const int*   dn = (const int*)d_in[3];    // dones     (B,S) i32
  float* partials = (float*)d_ws;           // 2048*4 floats, fully rewritten each call

  ppo_row_kernel<<<dim3(B_ROWS), dim3(TPB), 0, stream>>>(lp, rw, vl, dn, partials);
  ppo_finalize_kernel<<<dim3(1), dim3(TPB), 0, stream>>>(partials, (float*)d_out);
}